// RPNPostProcessor_773094113633
// MI455X (gfx1250) — compile-verified
//
#include <hip/hip_runtime.h>
#include <hip/hip_bf16.h>
#include <stdint.h>
#include <stddef.h>

// ---------------- configuration ----------------
#define NBINS          4096          // 12-bit radix histogram
#define BIN_SHIFT      20            // key >> 20 == top 12 bits
#define SEG            (NBINS / 256) // bins per thread in threshold scan
#define CAND_MAX       4096          // per-image candidate cap (>= 2000 + slack)
#define SORT_N         4096          // bitonic sort size (power of two)
#define PRE_NMS_TOP_N  2000
#define HIST_BLOCK     256
#define HIST_CHUNKS    128           // blocks per image for the histogram pass
#define BBOX_CLIP      4.135166556742356f  // log(1000/16)

// vector + address-space types for the async global->LDS builtin
typedef int v4i __attribute__((ext_vector_type(4)));
typedef __attribute__((address_space(1))) v4i gas_v4i;  // global
typedef __attribute__((address_space(3))) v4i las_v4i;  // LDS

// gfx1250 async global->LDS path (guarded: falls back to plain b128 loads)
#if defined(__gfx1250__) && __has_builtin(__builtin_amdgcn_global_load_async_to_lds_b128)
#define HAS_ASYNC_LDS 1
#else
#define HAS_ASYNC_LDS 0
#endif

__device__ __forceinline__ void wait_async_le1() {
#if __has_builtin(__builtin_amdgcn_s_wait_asynccnt)
  __builtin_amdgcn_s_wait_asynccnt(1);
#elif defined(__gfx1250__)
  asm volatile("s_wait_asynccnt 1" ::: "memory");
#endif
}
__device__ __forceinline__ void wait_async0() {
#if __has_builtin(__builtin_amdgcn_s_wait_asynccnt)
  __builtin_amdgcn_s_wait_asynccnt(0);
#elif defined(__gfx1250__)
  asm volatile("s_wait_asynccnt 0" ::: "memory");
#endif
}

// order-preserving float -> uint key (larger key == larger float)
__device__ __forceinline__ unsigned fkey(float f) {
  unsigned u = __float_as_uint(f);
  unsigned m = (u & 0x80000000u) ? 0xFFFFFFFFu : 0x80000000u;
  return u ^ m;
}
// inverse mapping
__device__ __forceinline__ float finv(unsigned k) {
  unsigned m = (k & 0x80000000u) ? 0x80000000u : 0xFFFFFFFFu;
  return __uint_as_float(k ^ m);
}

// ---------------- kernel 0: zero workspace ----------------
__global__ void rpn_init_kernel(unsigned* __restrict__ hist,
                                unsigned* __restrict__ counter,
                                unsigned* __restrict__ thresh,
                                int histLen, int n) {
  int i = blockIdx.x * blockDim.x + threadIdx.x;
  if (i < histLen) hist[i] = 0u;
  if (i < n) { counter[i] = 0u; thresh[i] = 0u; }
}

// ---------------- kernel 1: streaming radix histogram ----------------
// Double-buffered async global->LDS pipeline: tile t+1 is in flight on the
// async engine while tile t is histogrammed. ASYNCcnt completes in order, so
// s_wait_asynccnt 1 guarantees the older tile has landed in LDS.
__global__ void rpn_hist_kernel(const float* __restrict__ scores,
                                unsigned* __restrict__ ghist, int A) {
  __shared__ unsigned lhist[NBINS];
  __shared__ __align__(16) float stage[2][HIST_BLOCK * 4];
  const int b   = blockIdx.y;
  const int tid = threadIdx.x;
  const float* s = scores + (size_t)b * A;

  for (int i = tid; i < NBINS; i += HIST_BLOCK) lhist[i] = 0u;
  __syncthreads();

  const int chunk = (A + gridDim.x - 1) / gridDim.x;
  const int start = blockIdx.x * chunk;
  const int end   = min(start + chunk, A);
  const int tileElems = HIST_BLOCK * 4;
  const int span  = (end > start) ? (end - start) : 0;
  const int nTiles = span / tileElems;
  const int fullEnd = start + nTiles * tileElems;

#if HAS_ASYNC_LDS
  if (nTiles > 0) {
    const float* g0 = s + start + tid * 4;
    __builtin_amdgcn_global_load_async_to_lds_b128(
        (gas_v4i*)(float*)g0, (las_v4i*)&stage[0][tid * 4], 0, 0);
  }
  for (int ti = 0; ti < nTiles; ++ti) {
    if (ti + 1 < nTiles) {
      const float* gn = s + start + (ti + 1) * tileElems + tid * 4;
      __builtin_prefetch(gn + tileElems, 0, 1);   // global_prefetch_b8, 2 ahead
      __builtin_amdgcn_global_load_async_to_lds_b128(
          (gas_v4i*)(float*)gn, (las_v4i*)&stage[(ti + 1) & 1][tid * 4], 0, 0);
      wait_async_le1();   // oldest (tile ti) complete; next stays in flight
    } else {
      wait_async0();
    }
    float4 v = *(const float4*)&stage[ti & 1][tid * 4];
    atomicAdd(&lhist[fkey(v.x) >> BIN_SHIFT], 1u);
    atomicAdd(&lhist[fkey(v.y) >> BIN_SHIFT], 1u);
    atomicAdd(&lhist[fkey(v.z) >> BIN_SHIFT], 1u);
    atomicAdd(&lhist[fkey(v.w) >> BIN_SHIFT], 1u);
  }
#else
  for (int t = start; t < fullEnd; t += tileElems) {
    const float* g = s + t + tid * 4;
    __builtin_prefetch(g + tileElems, 0, 1);
    float4 v = *(const float4*)g;
    atomicAdd(&lhist[fkey(v.x) >> BIN_SHIFT], 1u);
    atomicAdd(&lhist[fkey(v.y) >> BIN_SHIFT], 1u);
    atomicAdd(&lhist[fkey(v.z) >> BIN_SHIFT], 1u);
    atomicAdd(&lhist[fkey(v.w) >> BIN_SHIFT], 1u);
  }
#endif
  for (int i = fullEnd + tid; i < end; i += HIST_BLOCK)
    atomicAdd(&lhist[fkey(s[i]) >> BIN_SHIFT], 1u);

  __syncthreads();
  unsigned* gh = ghist + (size_t)b * NBINS;
  for (int i = tid; i < NBINS; i += HIST_BLOCK)
    if (lhist[i]) atomicAdd(&gh[i], lhist[i]);
}

// ---------------- kernel 2: find per-image key threshold ----------------
__global__ void rpn_thresh_kernel(const unsigned* __restrict__ ghist,
                                  unsigned* __restrict__ thresh, int K) {
  __shared__ unsigned seg[256];
  __shared__ unsigned pre[257];
  const int b = blockIdx.x;
  const int t = threadIdx.x;
  const unsigned* h = ghist + (size_t)b * NBINS;

  // descending index d: d=0 is the highest bin (NBINS-1)
  unsigned sum = 0;
  for (int d = t * SEG; d < (t + 1) * SEG; ++d) sum += h[NBINS - 1 - d];
  seg[t] = sum;
  __syncthreads();
  if (t == 0) {
    unsigned run = 0;
    for (int i = 0; i < 256; ++i) { pre[i] = run; run += seg[i]; }
    pre[256] = run;
  }
  __syncthreads();
  if (pre[t] < (unsigned)K && pre[t + 1] >= (unsigned)K) {
    unsigned c = pre[t];
    for (int d = t * SEG; d < (t + 1) * SEG; ++d) {
      c += h[NBINS - 1 - d];
      if (c >= (unsigned)K) { thresh[b] = (unsigned)(NBINS - 1 - d) << BIN_SHIFT; break; }
    }
  }
  if (t == 0 && pre[256] < (unsigned)K) thresh[b] = 0u;  // degenerate: take all
}

// ---------------- kernel 3: compact candidates >= threshold ----------------
// Vectorized: 4 elements per lane via b128 loads; batch = blockIdx.y.
__global__ void rpn_compact_kernel(const float* __restrict__ scores,
                                   const unsigned* __restrict__ thresh,
                                   unsigned* __restrict__ counter,
                                   unsigned* __restrict__ ckeys,
                                   int* __restrict__ cidx,
                                   int A) {
  const int b  = blockIdx.y;
  const int i4 = (blockIdx.x * blockDim.x + threadIdx.x) * 4;
  if (i4 >= A) return;
  const float* s = scores + (size_t)b * A;
  const unsigned th = thresh[b];

  float v[4];
  int cnt;
  if (i4 + 3 < A) {
    float4 f = *(const float4*)(s + i4);
    v[0] = f.x; v[1] = f.y; v[2] = f.z; v[3] = f.w;
    cnt = 4;
  } else {
    cnt = A - i4;
    for (int j = 0; j < cnt; ++j) v[j] = s[i4 + j];
  }
#pragma unroll
  for (int j = 0; j < 4; ++j) {
    if (j >= cnt) break;
    unsigned key = fkey(v[j]);
    if (key >= th) {
      unsigned pos = atomicAdd(&counter[b], 1u);
      if (pos < CAND_MAX) {
        ckeys[(size_t)b * CAND_MAX + pos] = key;
        cidx [(size_t)b * CAND_MAX + pos] = i4 + j;
      }
    }
  }
}

// ---------------- kernel 4: bitonic sort (desc) + gather + decode ----------------
__global__ void rpn_sort_decode_kernel(const float* __restrict__ anchors,
                                       const float* __restrict__ breg,
                                       const unsigned* __restrict__ ckeys,
                                       const int* __restrict__ cidx,
                                       const unsigned* __restrict__ counter,
                                       float* __restrict__ out,
                                       int A, int K) {
  __shared__ unsigned skey[SORT_N];
  __shared__ int      sidx[SORT_N];
  const int b   = blockIdx.x;
  const int tid = threadIdx.x;
  const int nth = blockDim.x;
  const unsigned M = min(counter[b], (unsigned)CAND_MAX);

  for (int i = tid; i < SORT_N; i += nth) {
    if ((unsigned)i < M) {
      skey[i] = ckeys[(size_t)b * CAND_MAX + i];
      sidx[i] = cidx [(size_t)b * CAND_MAX + i];
    } else {
      skey[i] = 0u;   // smallest possible key (pads the tail)
      sidx[i] = 0;
    }
  }
  __syncthreads();

  // bitonic sort, descending by key
  for (int kk = 2; kk <= SORT_N; kk <<= 1) {
    for (int j = kk >> 1; j > 0; j >>= 1) {
      for (int i = tid; i < SORT_N; i += nth) {
        int ixj = i ^ j;
        if (ixj > i) {
          unsigned a = skey[i], c = skey[ixj];
          bool descBlock = ((i & kk) == 0);
          bool doSwap = descBlock ? (a < c) : (a > c);
          if (doSwap) {
            skey[i] = c;  skey[ixj] = a;
            int ti = sidx[i]; sidx[i] = sidx[ixj]; sidx[ixj] = ti;
          }
        }
      }
      __syncthreads();
    }
  }

  // gather + decode top-K (maskrcnn-benchmark BoxCoder, weights (1,1,1,1))
  const float4* anc4 = reinterpret_cast<const float4*>(anchors);
  const float4* reg4 = reinterpret_cast<const float4*>(breg);
  for (int i = tid; i < K; i += nth) {
    unsigned key = skey[i];
    int a = sidx[i];
    float score = finv(key);
    size_t row = (size_t)b * A + (size_t)a;
    float4 an = anc4[row];
    float4 rg = reg4[row];

    float w  = an.z - an.x + 1.0f;
    float h  = an.w - an.y + 1.0f;
    float cx = an.x + 0.5f * w;
    float cy = an.y + 0.5f * h;
    float dw = fminf(rg.z, BBOX_CLIP);
    float dh = fminf(rg.w, BBOX_CLIP);
    float px = rg.x * w + cx;
    float py = rg.y * h + cy;
    float pw = __expf(dw) * w;
    float ph = __expf(dh) * h;

    size_t o = ((size_t)b * K + (size_t)i) * 5;
    out[o + 0] = px - 0.5f * pw;
    out[o + 1] = py - 0.5f * ph;
    out[o + 2] = px + 0.5f * pw - 1.0f;
    out[o + 3] = py + 0.5f * ph - 1.0f;
    out[o + 4] = score;
  }
}

// ---------------- host launcher ----------------
extern "C" void kernel_launch(void* const* d_in, const int* in_sizes, int n_in,
                              void* d_out, int out_size, void* d_ws, size_t ws_size,
                              hipStream_t stream) {
  (void)n_in; (void)ws_size;
  const float* anchors    = (const float*)d_in[0];
  const float* objectness = (const float*)d_in[1];
  const float* breg       = (const float*)d_in[2];
  float* out = (float*)d_out;

  const int NA = in_sizes[1];                    // N * A (objectness has C==1)
  int N = out_size / (PRE_NMS_TOP_N * 5);
  if (N <= 0) N = 1;
  const int A = NA / N;
  const int K = (PRE_NMS_TOP_N < A) ? PRE_NMS_TOP_N : A;

  // carve aligned workspace
  uintptr_t p = (uintptr_t)d_ws;
  auto take = [&](size_t bytes) -> void* {
    uintptr_t r = (p + 255) & ~(uintptr_t)255;
    p = r + bytes;
    return (void*)r;
  };
  unsigned* hist    = (unsigned*)take((size_t)N * NBINS * sizeof(unsigned));
  unsigned* thresh  = (unsigned*)take((size_t)N * sizeof(unsigned));
  unsigned* counter = (unsigned*)take((size_t)N * sizeof(unsigned));
  unsigned* ckeys   = (unsigned*)take((size_t)N * CAND_MAX * sizeof(unsigned));
  int*      cidx    = (int*)     take((size_t)N * CAND_MAX * sizeof(int));

  const int histLen = N * NBINS;
  rpn_init_kernel<<<(histLen + 255) / 256, 256, 0, stream>>>(hist, counter, thresh,
                                                             histLen, N);
  rpn_hist_kernel<<<dim3(HIST_CHUNKS, N), HIST_BLOCK, 0, stream>>>(objectness, hist, A);
  rpn_thresh_kernel<<<N, 256, 0, stream>>>(hist, thresh, K);

  const int vecPerImg = (A + 4 * 256 - 1) / (4 * 256);
  rpn_compact_kernel<<<dim3(vecPerImg, N), 256, 0, stream>>>(
      objectness, thresh, counter, ckeys, cidx, A);

  rpn_sort_decode_kernel<<<N, 1024, 0, stream>>>(anchors, breg, ckeys, cidx,
                                                 counter, out, A, K);
}